// EntityAwareAttention_1898375544903
// MI455X (gfx1250) — compile-verified
//
#include <hip/hip_runtime.h>

typedef __attribute__((ext_vector_type(2))) float v2f;
typedef __attribute__((ext_vector_type(8))) float v8f;

namespace {
constexpr int kB = 4;
constexpr int kS = 4096;
constexpr int kH = 768;
constexpr int kE = 64;
constexpr int kNT = kH / 16;          // 48 n-tiles of 16 columns
constexpr int kStride = kH + 4;       // LDS row stride (floats): conflict-free B-frags
}

// LDS byte offset of a dynamic-shared pointer (VDST operand of async-to-LDS).
// Dynamic LDS starts right after static LDS (= groupstaticsize()).
__device__ __forceinline__ unsigned lds_byte_off(const void* p, const void* smem_base) {
  return __builtin_amdgcn_groupstaticsize() +
         (unsigned)((const char*)p - (const char*)smem_base);
}

__device__ __forceinline__ void wait_asynccnt0() {
#if __has_builtin(__builtin_amdgcn_s_wait_asynccnt)
  __builtin_amdgcn_s_wait_asynccnt(0);
#else
  asm volatile("s_wait_asynccnt 0x0" ::: "memory");
#endif
}

// ---------------------------------------------------------------------------
// Stage 1: Q/K/V projections.  out[m, n] = sum_h hs[m, h] * W[n, h] + b[n]
// One wave computes a 16x64 tile: the A-fragment (hs) is reused across 4
// B-fragments/accumulators -> 4x v_wmma_f32_16x16x4_f32 per K-chunk.
// ---------------------------------------------------------------------------
__global__ __launch_bounds__(32) void qkv_proj_kernel(
    const float* __restrict__ hs,
    const float* __restrict__ Wq, const float* __restrict__ bq,
    const float* __restrict__ Wk, const float* __restrict__ bk,
    const float* __restrict__ Wv, const float* __restrict__ bv,
    float* __restrict__ qo, float* __restrict__ ko, float* __restrict__ vo) {
  const float* W; const float* bias; float* out;
  if (blockIdx.z == 0)      { W = Wq; bias = bq; out = qo; }
  else if (blockIdx.z == 1) { W = Wk; bias = bk; out = ko; }
  else                      { W = Wv; bias = bv; out = vo; }

  const int m0   = blockIdx.x * 16;
  const int n0   = blockIdx.y * 64;
  const int lane = threadIdx.x & 31;
  const int half = lane >> 4;
  const int l16  = lane & 15;
  const int koff = half * 2;

  const float* arow = hs + (size_t)(m0 + l16) * kH + koff;
  const float* brow[4];
#pragma unroll
  for (int t = 0; t < 4; ++t)
    brow[t] = W + (size_t)(n0 + 16 * t + l16) * kH + koff;   // W[n,k] == B[k][n]

  v8f acc[4] = {};
#pragma unroll 4
  for (int c = 0; c < kH; c += 4) {
    v2f a = *(const v2f*)(arow + c);
#pragma unroll
    for (int t = 0; t < 4; ++t) {
      v2f b = *(const v2f*)(brow[t] + c);
      acc[t] = __builtin_amdgcn_wmma_f32_16x16x4_f32(false, a, false, b,
                                                     (short)0, acc[t], false, false);
    }
  }

#pragma unroll
  for (int t = 0; t < 4; ++t) {
    const float bn = bias[n0 + 16 * t + l16];
#pragma unroll
    for (int r = 0; r < 8; ++r) {
      const int row = r + half * 8;
      out[(size_t)(m0 + row) * kH + (n0 + 16 * t + l16)] = acc[t][r] + bn;
    }
  }
}

// ---------------------------------------------------------------------------
// Stage 2: per-key-column bias = (1 - mask) * -1e4, then +1.0 at each entity
// position (atomicAdd so duplicate positions accumulate, matching .at[].add).
// ---------------------------------------------------------------------------
__global__ void bias_init_kernel(const float* __restrict__ mask,
                                 float* __restrict__ biascol) {
  const int i = blockIdx.x * blockDim.x + threadIdx.x;
  if (i < kB * kS) biascol[i] = (1.0f - mask[i]) * -10000.0f;
}

__global__ void bias_scatter_kernel(const int* __restrict__ epos,
                                    float* __restrict__ biascol) {
  const int i = blockIdx.x * blockDim.x + threadIdx.x;
  if (i < kB * kE) {
    const int b = i / kE;
    const int p = epos[i];
    atomicAdd(&biascol[(size_t)b * kS + p], 1.0f);
  }
}

// ---------------------------------------------------------------------------
// Stage 3: fused flash attention.
//  * 256-thread workgroup = 8 waves = 128 query rows (8-way K/V sharing).
//  * K/V tiles (16x768 fp32) double-buffered in LDS, filled with
//    global_load_async_to_lds_b128 (ASYNCcnt path, no VGPR round-trip):
//    issue tile j+1 into the spare buffer, compute tile j, then one
//    s_wait_asynccnt 0 + one barrier per iteration.
//  * Scores: 4 independent WMMA accumulator chains (ILP) over LDS K frags;
//    online softmax via __shfl_xor inside 16-lane halves (wave32).
//  * Context accumulator: 48 x v8f = 384 VGPRs register-resident.
//  * LDS rows padded to 772 floats -> conflict-free B-fragment ds_reads.
// ---------------------------------------------------------------------------
__global__ __launch_bounds__(256, 1) void flash_attn_kernel(
    const float* __restrict__ q, const float* __restrict__ k,
    const float* __restrict__ v, const float* __restrict__ biascol,
    float* __restrict__ out) {
  extern __shared__ float smem[];
  float* sK0 = smem;
  float* sV0 = sK0 + 16 * kStride;
  float* sK1 = sV0 + 16 * kStride;
  float* sV1 = sK1 + 16 * kStride;
  float* sP  = sV1 + 16 * kStride;        // 8 waves * 256

  const int tid  = threadIdx.x;
  const int wave = tid >> 5;
  const int lane = tid & 31;
  const int half = lane >> 4;
  const int l16  = lane & 15;
  const int koff = half * 2;

  const int b  = blockIdx.y;
  const int m0 = blockIdx.x * 128 + wave * 16;    // this wave's query rows
  const float scale = 0.03608439182435161f;       // 1/sqrt(768)

  const float* qrow  = q + ((size_t)b * kS + m0 + l16) * kH + koff;
  const float* kbase = k + (size_t)b * kS * kH;
  const float* vbase = v + (size_t)b * kS * kH;
  const float* bcolp = biascol + (size_t)b * kS;
  float* myP = sP + wave * 256;

  // async-stage one 16-key K/V tile pair into (dK, dV)
  auto stage = [&](float* dK, float* dV, int j0) {
    const float* gK = kbase + (size_t)j0 * kH;
    const float* gV = vbase + (size_t)j0 * kH;
#pragma unroll 4
    for (int idx = tid; idx < 16 * (kH / 4); idx += 256) {
      const int row = idx / (kH / 4);
      const int col = idx - row * (kH / 4);
      const unsigned lK = lds_byte_off(dK + row * kStride + col * 4, smem);
      const unsigned lV = lds_byte_off(dV + row * kStride + col * 4, smem);
      const float* pK = gK + idx * 4;
      const float* pV = gV + idx * 4;
      asm volatile("global_load_async_to_lds_b128 %0, %1, off"
                   :: "v"(lK), "v"(pK) : "memory");
      asm volatile("global_load_async_to_lds_b128 %0, %1, off"
                   :: "v"(lV), "v"(pV) : "memory");
    }
  };

  v8f acc[kNT] = {};
  float Mst[8], Lst[8];
#pragma unroll
  for (int r = 0; r < 8; ++r) { Mst[r] = -1e30f; Lst[r] = 0.0f; }

  // prologue: fill buffer 0 with the first key tile
  stage(sK0, sV0, 0);
  wait_asynccnt0();
  __syncthreads();

  float* curK = sK0; float* curV = sV0;
  float* nxtK = sK1; float* nxtV = sV1;

  for (int j0 = 0; j0 < kS; j0 += 16) {
    // prefetch next tile into the spare buffer while we compute this one
    if (j0 + 16 < kS) stage(nxtK, nxtV, j0 + 16);

    // ---- S = Q K^T : 4 independent accumulator chains ----
    v8f s4[4] = {};
#pragma unroll 2
    for (int c = 0; c < kH; c += 16) {
#pragma unroll
      for (int u = 0; u < 4; ++u) {
        v2f a  = *(const v2f*)(qrow + c + 4 * u);                       // L2 Q
        v2f bf = *(const v2f*)(&curK[l16 * kStride + c + 4 * u + koff]); // LDS K
        s4[u] = __builtin_amdgcn_wmma_f32_16x16x4_f32(false, a, false, bf,
                                                      (short)0, s4[u], false, false);
      }
    }
    v8f s = (s4[0] + s4[1]) + (s4[2] + s4[3]);

    // ---- scale + key-column bias + online softmax bookkeeping ----
    const float bc = bcolp[j0 + l16];   // column = l16 for every row r
    float alpha[8];
#pragma unroll
    for (int r = 0; r < 8; ++r) {
      float sv = s[r] * scale + bc;
      float m = sv;
      m = fmaxf(m, __shfl_xor(m, 1, 32));
      m = fmaxf(m, __shfl_xor(m, 2, 32));
      m = fmaxf(m, __shfl_xor(m, 4, 32));
      m = fmaxf(m, __shfl_xor(m, 8, 32));      // row max within 16-lane half
      const float newM = fmaxf(Mst[r], m);
      const float a_r  = __expf(Mst[r] - newM);
      const float p    = __expf(sv - newM);
      float ps = p;
      ps += __shfl_xor(ps, 1, 32);
      ps += __shfl_xor(ps, 2, 32);
      ps += __shfl_xor(ps, 4, 32);
      ps += __shfl_xor(ps, 8, 32);             // row sum
      Lst[r]   = Lst[r] * a_r + ps;
      Mst[r]   = newM;
      alpha[r] = a_r;
      myP[(r + half * 8) * 16 + l16] = p;      // row-major P tile (wave-private)
    }

    // P as A-fragments: chunk kc covers K = kc*4 .. kc*4+3 (in-wave LDS order)
    v2f pa[4];
#pragma unroll
    for (int kc = 0; kc < 4; ++kc)
      pa[kc] = *(const v2f*)(&myP[l16 * 16 + kc * 4 + koff]);

    // ---- context = alpha*context + P @ V  (register accumulator) ----
#pragma unroll
    for (int nt = 0; nt < kNT; ++nt) {
      const int n0 = nt * 16;
      v8f c0 = acc[nt];
#pragma unroll
      for (int r = 0; r < 8; ++r) c0[r] *= alpha[r];
#pragma unroll
      for (int kc = 0; kc < 4; ++kc) {
        const int kk = kc * 4 + koff;          // key row inside the tile
        v2f vf;
        vf.x = curV[kk * kStride + n0 + l16];
        vf.y = curV[(kk + 1) * kStride + n0 + l16];
        c0 = __builtin_amdgcn_wmma_f32_16x16x4_f32(false, pa[kc], false, vf,
                                                   (short)0, c0, false, false);
      }
      acc[nt] = c0;
    }

    // next tile fully landed (own issues done) + all waves done reading cur
    wait_asynccnt0();
    __syncthreads();
    float* t0 = curK; curK = nxtK; nxtK = t0;
    float* t1 = curV; curV = nxtV; nxtV = t1;
  }

  // ---- normalize by row sum and write out ----
  float rL[8];
#pragma unroll
  for (int r = 0; r < 8; ++r) rL[r] = 1.0f / Lst[r];
  float* orow = out + ((size_t)b * kS + m0) * kH;
#pragma unroll
  for (int nt = 0; nt < kNT; ++nt) {
    const int n0 = nt * 16;
#pragma unroll
    for (int r = 0; r < 8; ++r) {
      const int row = r + half * 8;
      orow[(size_t)row * kH + n0 + l16] = acc[nt][r] * rL[r];
    }
  }
}

// ---------------------------------------------------------------------------
extern "C" void kernel_launch(void* const* d_in, const int* in_sizes, int n_in,
                              void* d_out, int out_size, void* d_ws, size_t ws_size,
                              hipStream_t stream) {
  const float* hs   = (const float*)d_in[0];
  const float* mask = (const float*)d_in[1];
  const int*   epos = (const int*)d_in[2];
  const float* Wq   = (const float*)d_in[3];
  const float* bq   = (const float*)d_in[4];
  const float* Wk   = (const float*)d_in[5];
  const float* bk   = (const float*)d_in[6];
  const float* Wv   = (const float*)d_in[7];
  const float* bv   = (const float*)d_in[8];
  float* outp = (float*)d_out;

  const size_t nBSH = (size_t)kB * kS * kH;
  float* qbuf = (float*)d_ws;
  float* kbuf = qbuf + nBSH;
  float* vbuf = kbuf + nBSH;
  float* bias = vbuf + nBSH;

  qkv_proj_kernel<<<dim3(kB * kS / 16, kH / 64, 3), 32, 0, stream>>>(
      hs, Wq, bq, Wk, bk, Wv, bv, qbuf, kbuf, vbuf);
  bias_init_kernel<<<(kB * kS + 255) / 256, 256, 0, stream>>>(mask, bias);
  bias_scatter_kernel<<<1, kB * kE, 0, stream>>>(epos, bias);

  const size_t smemBytes = (size_t)(4 * 16 * kStride + 8 * 256) * sizeof(float);
  flash_attn_kernel<<<dim3(kS / 128, kB), 256, smemBytes, stream>>>(
      qbuf, kbuf, vbuf, bias, outp);
}